// SSDLoss_39779987095893
// MI455X (gfx1250) — compile-verified
//
#include <hip/hip_runtime.h>
#include <cstdint>

// ---------------- problem constants ----------------
#define NUM_CLASSES 21
#define BN          (64 * 24564)        // 1,572,096 boxes total
#define BPB         256                 // boxes per block
#define NBLK        (BN / BPB)          // 6141 (exact)
#define ROWB        (NUM_CLASSES * 4)   // 84 bytes per box row
#define TILE_DW     (BPB * NUM_CLASSES) // 5376 dwords staged per block (21504 B)
#define H1          8192                // radix level 1: float bits[31:18]
#define H2          16384               // radix level 2: float bits[17:4]

// ---------------- workspace layout (4-byte element offsets) ----------------
#define OFF_NEG 0                        // float[BN]  neg CE values (-1 for positives)
#define OFF_H1  (OFF_NEG + BN)           // int[H1]
#define OFF_H2  (OFF_H1 + H1)            // int[H2]
#define OFF_PPC (OFF_H2 + H2)            // float[NBLK] partial pos-CE sums
#define OFF_PLS (OFF_PPC + NBLK)         // float[NBLK] partial smooth-L1 sums
#define OFF_PNP (OFF_PLS + NBLK)         // int[NBLK]   partial pos counts
#define OFF_PNS (OFF_PNP + NBLK)         // float[NBLK] partial neg-loss sums
#define OFF_SC  (OFF_PNS + NBLK)         // scalars[16]
// scalars: 0 numPos(i) 1 posCeSum(f) 2 locSum(f) 3 numNeg(i) 4 thr1(i) 5 k1(i) 6 thr2(i) 7 k2(i)

typedef unsigned int u32x4 __attribute__((ext_vector_type(4)));
typedef int          i32x4 __attribute__((ext_vector_type(4)));
typedef int          i32x8 __attribute__((ext_vector_type(8)));

// ---------------- kernels ----------------
__global__ __launch_bounds__(256) void k_zero(float* ws) {
    int i = blockIdx.x * 256 + threadIdx.x;
    int* wsI = (int*)ws;
    if (i < H1 + H2)            wsI[OFF_H1 + i] = 0;
    else if (i < H1 + H2 + 16)  wsI[OFF_SC + (i - H1 - H2)] = 0;
}

// Pass 1: TDM (Tensor Data Mover) stages each block's contiguous 21504-byte
// cls tile into LDS with ONE instruction; then per-box CE, level-1 radix
// histogram, and deterministic per-block partials.
__global__ __launch_bounds__(256) void k_ce(const float* __restrict__ locp,
                                            const float* __restrict__ clsp,
                                            const int*   __restrict__ tgt,
                                            const float* __restrict__ loct,
                                            float* __restrict__ ws) {
    __shared__ __align__(16) float stage[TILE_DW];   // 21504 B
    __shared__ float sF[256];
    __shared__ float sG[256];
    __shared__ int   sI[256];
    int* wsI = (int*)ws;
    const int t = threadIdx.x, blk = blockIdx.x;

    // --- TDM: one tensor_load_to_lds per block (wave 0 lane 0 issues it) ---
    if (t == 0) {
        unsigned long long ga =
            (unsigned long long)((const char*)clsp + (size_t)blk * (size_t)(BPB * ROWB));
        unsigned l0 = (unsigned)(uintptr_t)(&stage[0]);  // low 32 bits = LDS offset

        u32x4 g0;
        g0.x = 1u;                                       // count=1 (valid user D#)
        g0.y = l0;                                       // lds_addr
        g0.z = (unsigned)(ga & 0xFFFFFFFFu);             // global_addr[31:0]
        g0.w = (unsigned)((ga >> 32) & 0x01FFFFFFu)      // global_addr[56:32]
             | (2u << 30);                               // type = 2 ("image")

        i32x8 g1;
        g1[0] = 0x00020000;                  // wg_mask=0, data_size=2 (4 bytes)
        g1[1] = (int)((unsigned)TILE_DW << 16);  // tensor_dim0[15:0] @ bits[63:48]
        g1[2] = (int)(1u << 16);             // tensor_dim0[31:16]=0 ; tensor_dim1=1
        g1[3] = (int)((unsigned)TILE_DW << 16);  // tile_dim0 = 5376 @ bits[127:112]
        g1[4] = 0;                           // tile_dim1 = 0 (1D), tile_dim2 = 0
        g1[5] = TILE_DW;                     // tensor_dim0_stride[31:0]
        g1[6] = 0;                           // stride hi / dim1_stride lo
        g1[7] = 0;                           // dim1_stride hi

        i32x4 gz = {0, 0, 0, 0};
#if defined(__clang_major__) && (__clang_major__ >= 23)
        i32x8 gz8 = {0, 0, 0, 0, 0, 0, 0, 0};
        __builtin_amdgcn_tensor_load_to_lds(g0, g1, gz, gz, gz8, 0);
#else
        __builtin_amdgcn_tensor_load_to_lds(g0, g1, gz, gz, 0);
#endif
        __builtin_amdgcn_s_wait_tensorcnt(0);
    }
    __syncthreads();

    // --- per-box CE (LDS reads: stride 21 dwords, gcd(21,64)=1 -> conflict-free) ---
    const int    box = blk * BPB + t;
    const float* row = &stage[t * NUM_CLASSES];
    float m = row[0];
#pragma unroll
    for (int c = 1; c < NUM_CLASSES; ++c) m = fmaxf(m, row[c]);
    float s = 0.f;
#pragma unroll
    for (int c = 0; c < NUM_CLASSES; ++c) s += __expf(row[c] - m);
    float lse = __logf(s) + m;

    int   tg  = tgt[box];
    bool  pos = tg > 0;
    float ce  = fmaxf(lse - row[tg], 0.f);   // CE >= 0 (guard fp round)

    ws[OFF_NEG + box] = pos ? -1.0f : ce;
    if (!pos) atomicAdd(&wsI[OFF_H1 + (__float_as_uint(ce) >> 18)], 1);

    float pc = pos ? ce : 0.f, lsum = 0.f;
    int   np = pos ? 1 : 0;
    if (pos) {   // smooth-L1 only for ~2% positives
        const float4 p = ((const float4*)locp)[box];
        const float4 q = ((const float4*)loct)[box];
        float d0 = p.x - q.x, d1 = p.y - q.y, d2 = p.z - q.z, d3 = p.w - q.w;
        float a0 = fabsf(d0), a1 = fabsf(d1), a2 = fabsf(d2), a3 = fabsf(d3);
        lsum = (a0 < 1.f ? 0.5f * d0 * d0 : a0 - 0.5f)
             + (a1 < 1.f ? 0.5f * d1 * d1 : a1 - 0.5f)
             + (a2 < 1.f ? 0.5f * d2 * d2 : a2 - 0.5f)
             + (a3 < 1.f ? 0.5f * d3 * d3 : a3 - 0.5f);
    }

    sF[t] = pc; sG[t] = lsum; sI[t] = np;
    __syncthreads();
    for (int o = 128; o > 0; o >>= 1) {
        if (t < o) { sF[t] += sF[t + o]; sG[t] += sG[t + o]; sI[t] += sI[t + o]; }
        __syncthreads();
    }
    if (t == 0) {
        ws[OFF_PPC + blk]  = sF[0];
        ws[OFF_PLS + blk]  = sG[0];
        wsI[OFF_PNP + blk] = sI[0];
    }
}

// Single block: deterministic reduce of partials, then level-1 threshold scan.
__global__ __launch_bounds__(256) void k_select1(float* ws) {
    int* wsI = (int*)ws;
    const int t = threadIdx.x;
    __shared__ float sF[256]; __shared__ float sG[256]; __shared__ int sI[256];
    __shared__ int numNegS;

    float pc = 0.f, ls = 0.f; int np = 0;
    for (int i = t; i < NBLK; i += 256) {
        pc += ws[OFF_PPC + i]; ls += ws[OFF_PLS + i]; np += wsI[OFF_PNP + i];
    }
    sF[t] = pc; sG[t] = ls; sI[t] = np;
    __syncthreads();
    for (int o = 128; o > 0; o >>= 1) {
        if (t < o) { sF[t] += sF[t + o]; sG[t] += sG[t + o]; sI[t] += sI[t + o]; }
        __syncthreads();
    }
    if (t == 0) {
        int numPos = sI[0];
        wsI[OFF_SC + 0] = numPos;
        ws [OFF_SC + 1] = sF[0];
        ws [OFF_SC + 2] = sG[0];
        int numNeg = min(3 * numPos, BN - numPos);
        wsI[OFF_SC + 3] = numNeg;
        numNegS = numNeg;
    }
    __syncthreads();
    const int numNeg = numNegS;

    // top-down scan of hist1: thread t owns 32 bins, chunk 0 = highest bins
    int base = H1 - 32 * (t + 1), cs = 0;
    for (int j = 0; j < 32; ++j) cs += wsI[OFF_H1 + base + j];
    sI[t] = cs;
    __syncthreads();
    if (t == 0) {
        int thr, k;
        if (numNeg == 0) { thr = 0x7fffffff; k = 0; }
        else {
            long cum = 0; int c = 0;
            for (; c < 256; ++c) { if (cum + sI[c] >= numNeg) break; cum += sI[c]; }
            if (c == 256) { thr = -1; k = 0; }     // take all negatives
            else {
                int b2 = H1 - 32 * (c + 1); thr = 0x7fffffff; k = 0;
                for (int b = 31; b >= 0; --b) {
                    int h = wsI[OFF_H1 + b2 + b];
                    if (cum + h >= numNeg) { thr = b2 + b; k = numNeg - (int)cum; break; }
                    cum += h;
                }
            }
        }
        wsI[OFF_SC + 4] = thr; wsI[OFF_SC + 5] = k;
    }
}

// Level-2 histogram over values inside the level-1 threshold bin (L2-resident pass).
__global__ __launch_bounds__(256) void k_hist2(float* ws) {
    int* wsI = (int*)ws;
    const int thr1 = wsI[OFF_SC + 4];
    const int i = blockIdx.x * 256 + threadIdx.x;   // grid covers BN exactly
    float v = ws[OFF_NEG + i];
    if (v >= 0.f) {
        unsigned key = __float_as_uint(v);
        if ((int)(key >> 18) == thr1)
            atomicAdd(&wsI[OFF_H2 + ((key >> 4) & 0x3FFF)], 1);
    }
}

__global__ __launch_bounds__(256) void k_select2(float* ws) {
    int* wsI = (int*)ws;
    const int t = threadIdx.x;
    __shared__ int sC[256];
    const int k1 = wsI[OFF_SC + 5];

    int base = H2 - 64 * (t + 1), cs = 0;
    for (int j = 0; j < 64; ++j) cs += wsI[OFF_H2 + base + j];
    sC[t] = cs;
    __syncthreads();
    if (t == 0) {
        int thr, k;
        if (k1 == 0) { thr = 0x7fffffff; k = 0; }
        else {
            long cum = 0; int c = 0;
            for (; c < 256; ++c) { if (cum + sC[c] >= k1) break; cum += sC[c]; }
            if (c == 256) { thr = -1; k = 0; }
            else {
                int b2 = H2 - 64 * (c + 1); thr = 0x7fffffff; k = 0;
                for (int b = 63; b >= 0; --b) {
                    int h = wsI[OFF_H2 + b2 + b];
                    if (cum + h >= k1) { thr = b2 + b; k = k1 - (int)cum; break; }
                    cum += h;
                }
            }
        }
        wsI[OFF_SC + 6] = thr; wsI[OFF_SC + 7] = k;
    }
}

// Masked sum of negatives strictly above the refined threshold.
__global__ __launch_bounds__(256) void k_negsum(float* ws) {
    int* wsI = (int*)ws;
    const int thr1 = wsI[OFF_SC + 4], thr2 = wsI[OFF_SC + 6];
    const int i = blockIdx.x * 256 + threadIdx.x;
    float v = ws[OFF_NEG + i], s = 0.f;
    if (v >= 0.f) {
        unsigned key = __float_as_uint(v);
        int b1 = (int)(key >> 18);
        if (b1 > thr1) s = v;
        else if (b1 == thr1 && (int)((key >> 4) & 0x3FFF) > thr2) s = v;
    }
    __shared__ float sF[256];
    sF[threadIdx.x] = s;
    __syncthreads();
    for (int o = 128; o > 0; o >>= 1) {
        if (threadIdx.x < o) sF[threadIdx.x] += sF[threadIdx.x + o];
        __syncthreads();
    }
    if (threadIdx.x == 0) ws[OFF_PNS + blockIdx.x] = sF[0];
}

__global__ __launch_bounds__(256) void k_final(float* ws, float* out) {
    int* wsI = (int*)ws;
    const int t = threadIdx.x;
    float s = 0.f;
    for (int i = t; i < NBLK; i += 256) s += ws[OFF_PNS + i];
    __shared__ float sF[256];
    sF[t] = s;
    __syncthreads();
    for (int o = 128; o > 0; o >>= 1) {
        if (t < o) sF[t] += sF[t + o];
        __syncthreads();
    }
    if (t == 0) {
        int   numPos = wsI[OFF_SC + 0];
        float posCe  = ws [OFF_SC + 1];
        float locSum = ws [OFF_SC + 2];
        int   thr1 = wsI[OFF_SC + 4], thr2 = wsI[OFF_SC + 6], k2 = wsI[OFF_SC + 7];
        float negSum = sF[0];
        if (k2 > 0 && thr1 >= 0 && thr1 < H1 && thr2 >= 0 && thr2 < H2)
            negSum += (float)k2 *
                      __uint_as_float(((unsigned)thr1 << 18) | ((unsigned)thr2 << 4));
        float cls = posCe + negSum;
        float loc = (numPos == 0) ? 0.f : locSum / fmaxf((float)numPos, 1.f);
        out[0] = cls + loc;   // total_loss (ALPHA = 1)
        out[1] = cls;         // cls_loss
        out[2] = loc;         // loc_loss
    }
}

// ---------------- host entry ----------------
extern "C" void kernel_launch(void* const* d_in, const int* in_sizes, int n_in,
                              void* d_out, int out_size, void* d_ws, size_t ws_size,
                              hipStream_t stream) {
    const float* locp = (const float*)d_in[0];   // [B,N,4]
    const float* clsp = (const float*)d_in[1];   // [B,N,21]
    const int*   tgt  = (const int*)  d_in[2];   // [B,N]
    const float* loct = (const float*)d_in[3];   // [B,N,4]
    float* ws  = (float*)d_ws;
    float* out = (float*)d_out;
    (void)in_sizes; (void)n_in; (void)out_size; (void)ws_size;

    k_zero   <<<(H1 + H2 + 16 + 255) / 256, 256, 0, stream>>>(ws);
    k_ce     <<<NBLK, 256, 0, stream>>>(locp, clsp, tgt, loct, ws);
    k_select1<<<1,    256, 0, stream>>>(ws);
    k_hist2  <<<NBLK, 256, 0, stream>>>(ws);
    k_select2<<<1,    256, 0, stream>>>(ws);
    k_negsum <<<NBLK, 256, 0, stream>>>(ws);
    k_final  <<<1,    256, 0, stream>>>(ws, out);
}